// PNAEPCSAFT_8126078124760
// MI455X (gfx1250) — compile-verified
//
#include <hip/hip_runtime.h>
#include <math.h>
#include <float.h>

typedef float v2f __attribute__((ext_vector_type(2)));
typedef float v8f __attribute__((ext_vector_type(8)));

#define N_NODES  20000
#define N_EDGES  60000
#define N_GRAPHS 512
#define TOWERS   4

// ---------------- float atomic min/max via int monotonic trick ----------------
__device__ __forceinline__ void atomicMaxF(float* addr, float val) {
  if (val >= 0.0f) atomicMax((int*)addr, __float_as_int(val));
  else             atomicMin((unsigned int*)addr, __float_as_uint(val));
}
__device__ __forceinline__ void atomicMinF(float* addr, float val) {
  if (val >= 0.0f) atomicMin((int*)addr, __float_as_int(val));
  else             atomicMax((unsigned int*)addr, __float_as_uint(val));
}

// ---------------- generic f32 WMMA GEMM: C = A[M,K] @ B[K,N] (+bias, act) -----
// one wave (32 lanes) computes a 16x16 C tile using V_WMMA_F32_16X16X4_F32.
// Loads in the hot loop are UNCONDITIONAL (clamped addresses) with cndmask
// zeroing, so no exec-mask branching / loadcnt drains around the WMMAs.
__global__ void gemm_f32_wmma(const float* __restrict__ A, int lda,
                              const float* __restrict__ B, int ldb,
                              const float* __restrict__ bias,
                              float* __restrict__ C, int ldc,
                              int M, int N, int K, int act) {
  const int lane = threadIdx.x & 31;
  const int m0 = blockIdx.x * 16;
  const int n0 = blockIdx.y * 16;
  const int r    = lane & 15;
  const int koff = (lane >> 4) << 1;   // lanes 0-15 -> K 0,1 ; lanes 16-31 -> K 2,3
  const int arow = m0 + r;
  const int bcol = n0 + r;
  const bool aok = (arow < M);
  const bool bok = (bcol < N);
  const size_t abase = (size_t)(aok ? arow : (M - 1)) * lda;
  const int    bc    = bok ? bcol : (N - 1);
  const float  am    = aok ? 1.f : 0.f;
  const float  bm    = bok ? 1.f : 0.f;

  v8f acc = {0.f, 0.f, 0.f, 0.f, 0.f, 0.f, 0.f, 0.f};
  const int Kfull = K & ~3;
  for (int k0 = 0; k0 < Kfull; k0 += 4) {
    const int ka = k0 + koff;          // ka, ka+1 < Kfull <= K always
    v2f a, b;
    a.x = A[abase + ka] * am;
    a.y = A[abase + ka + 1] * am;
    b.x = B[(size_t)ka * ldb + bc] * bm;
    b.y = B[(size_t)(ka + 1) * ldb + bc] * bm;
    acc = __builtin_amdgcn_wmma_f32_16x16x4_f32(false, a, false, b,
                                                (short)0, acc, false, false);
  }
  if (Kfull < K) {                      // K % 4 tail, branchless (clamp + mask)
    const int ka = Kfull + koff, kb = ka + 1;
    const int kac = (ka < K) ? ka : (K - 1);
    const int kbc = (kb < K) ? kb : (K - 1);
    const float ax = A[abase + kac];
    const float ay = A[abase + kbc];
    const float bx = B[(size_t)kac * ldb + bc];
    const float by = B[(size_t)kbc * ldb + bc];
    v2f a, b;
    a.x = (aok && ka < K) ? ax : 0.f;
    a.y = (aok && kb < K) ? ay : 0.f;
    b.x = (bok && ka < K) ? bx : 0.f;
    b.y = (bok && kb < K) ? by : 0.f;
    acc = __builtin_amdgcn_wmma_f32_16x16x4_f32(false, a, false, b,
                                                (short)0, acc, false, false);
  }
  if (bok) {
    const float bv = bias ? bias[bcol] : 0.f;
    const int rbase = m0 + ((lane >> 4) << 3);   // upper lanes: rows M+8..M+15
    #pragma unroll
    for (int i = 0; i < 8; ++i) {
      const int row = rbase + i;
      if (row < M) {
        float v = acc[i] + bv;
        if (act == 1) v = fmaxf(v, 0.f);
        else if (act == 2) v = tanhf(v);
        C[(size_t)row * ldc + bcol] = v;
      }
    }
  }
}

// -------- edge gather: H[e, 0:F]=x[dst], [F:2F]=x[src], [2F:3F]=edge encoder --
__global__ void gather_edges(const float* __restrict__ x, int F,
                             const float* __restrict__ ea,
                             const int* __restrict__ src,
                             const int* __restrict__ dst,
                             const float* __restrict__ ew,   // [3,F]
                             const float* __restrict__ eb,   // [F]
                             float* __restrict__ H) {        // [E, 3F]
  const int K3 = 3 * F;
  const int total = N_EDGES * K3;
  for (int idx = blockIdx.x * blockDim.x + threadIdx.x; idx < total;
       idx += gridDim.x * blockDim.x) {
    const int e = idx / K3;
    const int f3 = idx % K3;
    float v;
    if (f3 < F) v = x[(size_t)dst[e] * F + f3];
    else if (f3 < 2 * F) v = x[(size_t)src[e] * F + (f3 - F)];
    else {
      const int f = f3 - 2 * F;
      v = eb[f] + ea[e * 3 + 0] * ew[0 * F + f]
                + ea[e * 3 + 1] * ew[1 * F + f]
                + ea[e * 3 + 2] * ew[2 * F + f];
    }
    H[idx] = v;
  }
}

// -------- repack pre-MLP weights [T,3F,F] -> [3F, T*F] for one flat GEMM -----
__global__ void pack_prw(const float* __restrict__ prw, float* __restrict__ Wpk, int F) {
  const int K3 = 3 * F, TF = TOWERS * F;
  const int total = K3 * TF;
  for (int idx = blockIdx.x * blockDim.x + threadIdx.x; idx < total;
       idx += gridDim.x * blockDim.x) {
    const int f = idx / TF;
    const int to = idx % TF;
    const int t = to / F, o = to % F;
    Wpk[idx] = prw[((size_t)t * K3 + f) * F + o];
  }
}

// -------- aggregate buffers init -----------------------------------------------
__global__ void fill_agg(float* s1, float* s2, float* mn, float* mx, float* deg, int ntf) {
  for (int i = blockIdx.x * blockDim.x + threadIdx.x; i < ntf;
       i += gridDim.x * blockDim.x) {
    s1[i] = 0.f; s2[i] = 0.f; mn[i] = FLT_MAX; mx[i] = -FLT_MAX;
    if (i < N_NODES) deg[i] = 0.f;
  }
}
__global__ void fill_zero(float* p, int n) {
  for (int i = blockIdx.x * blockDim.x + threadIdx.x; i < n;
       i += gridDim.x * blockDim.x) p[i] = 0.f;
}

__global__ void deg_kernel(const int* __restrict__ dst, float* __restrict__ deg) {
  const int e = blockIdx.x * blockDim.x + threadIdx.x;
  if (e < N_EDGES) atomicAdd(&deg[dst[e]], 1.f);
}

// -------- scatter msg -> sum / sumsq / min / max --------------------------------
__global__ void scatter_agg(const float* __restrict__ msg, const int* __restrict__ dst,
                            float* s1, float* s2, float* mn, float* mx, int TF) {
  const int total = N_EDGES * TF;
  for (int idx = blockIdx.x * blockDim.x + threadIdx.x; idx < total;
       idx += gridDim.x * blockDim.x) {
    const int e = idx / TF;
    const int c = idx % TF;
    const float v = msg[idx];
    const size_t o = (size_t)dst[e] * TF + c;
    atomicAdd(&s1[o], v);
    atomicAdd(&s2[o], v * v);
    atomicMaxF(&mx[o], v);
    atomicMinF(&mn[o], v);
  }
}

// -------- fused PNA scalers + per-tower post GEMM (comb tile built in LDS) ------
// grid: (ceil(N/16), TOWERS), block: 64 (2 waves: output col tiles 0-15, 16-31)
__global__ void pna_post(const float* __restrict__ h, int F,
                         const float* __restrict__ s1, const float* __restrict__ s2,
                         const float* __restrict__ mn, const float* __restrict__ mx,
                         const float* __restrict__ deg,
                         const float* __restrict__ powW,  // [T, 13F, 18]
                         const float* __restrict__ pob,   // [T, 18]
                         float* __restrict__ out,         // [N, T*18]
                         float avg_log) {
  extern __shared__ float sm[];                 // [16 x 13F]
  const int F13 = 13 * F;
  const int TF  = TOWERS * F;
  const int t = blockIdx.y;
  const int node0 = blockIdx.x * 16;

  for (int idx = threadIdx.x; idx < 16 * F13; idx += blockDim.x) {
    const int r = idx / F13, ft = idx % F13;
    const int n = node0 + r;
    float v = 0.f;
    if (n < N_NODES) {
      if (ft < F) v = h[(size_t)n * F + ft];
      else {
        const int c = (ft - F) / F;             // chunk 0..11
        const int f = (ft - F) % F;
        const int b = c & 3;                    // 0=mean 1=min 2=max 3=std
        const int s = c >> 2;                   // 0=id 1=amp 2=att
        const size_t o = (size_t)n * TF + t * F + f;
        const float d  = deg[n];
        const float dc = fmaxf(d, 1.f);
        float base;
        if (b == 0) base = s1[o] / dc;
        else if (b == 1) base = (d > 0.f) ? mn[o] : 0.f;
        else if (b == 2) base = (d > 0.f) ? mx[o] : 0.f;
        else {
          const float mean = s1[o] / dc;
          const float var  = s2[o] / dc - mean * mean;
          base = sqrtf(fmaxf(var, 0.f) + 1e-5f);
        }
        const float amp = logf(dc + 1.f) / avg_log;
        const float fac = (s == 0) ? 1.f : ((s == 1) ? amp : 1.f / amp);
        v = base * fac;
      }
    }
    sm[idx] = v;
  }
  __syncthreads();

  const int lane = threadIdx.x & 31;
  const int wave = threadIdx.x >> 5;
  const int rr   = lane & 15;
  const int koff = (lane >> 4) << 1;
  const int bcol = wave * 16 + rr;
  const bool bok = (bcol < 18);
  const int  bc  = bok ? bcol : 17;
  const float bmask = bok ? 1.f : 0.f;
  const float* Bw = powW + (size_t)t * F13 * 18;

  v8f acc = {0.f, 0.f, 0.f, 0.f, 0.f, 0.f, 0.f, 0.f};
  const int Kfull = F13 & ~3;
  for (int k0 = 0; k0 < Kfull; k0 += 4) {
    const int ka = k0 + koff;
    v2f a, b;
    a.x = sm[rr * F13 + ka];
    a.y = sm[rr * F13 + ka + 1];
    b.x = Bw[(size_t)ka * 18 + bc] * bmask;
    b.y = Bw[(size_t)(ka + 1) * 18 + bc] * bmask;
    acc = __builtin_amdgcn_wmma_f32_16x16x4_f32(false, a, false, b,
                                                (short)0, acc, false, false);
  }
  if (Kfull < F13) {                    // F13 % 4 tail (F13=117 for layer 0)
    const int ka = Kfull + koff, kb = ka + 1;
    const int kac = (ka < F13) ? ka : (F13 - 1);
    const int kbc = (kb < F13) ? kb : (F13 - 1);
    const float ax = sm[rr * F13 + kac];
    const float ay = sm[rr * F13 + kbc];
    const float bx = Bw[(size_t)kac * 18 + bc];
    const float by = Bw[(size_t)kbc * 18 + bc];
    v2f a, b;
    a.x = (ka < F13) ? ax : 0.f;
    a.y = (kb < F13) ? ay : 0.f;
    b.x = (bok && ka < F13) ? bx : 0.f;
    b.y = (bok && kb < F13) ? by : 0.f;
    acc = __builtin_amdgcn_wmma_f32_16x16x4_f32(false, a, false, b,
                                                (short)0, acc, false, false);
  }
  if (bok) {
    const float bv = pob[t * 18 + bcol];
    const int rbase = (lane >> 4) << 3;
    #pragma unroll
    for (int i = 0; i < 8; ++i) {
      const int n = node0 + rbase + i;
      if (n < N_NODES)
        out[(size_t)n * (TOWERS * 18) + t * 18 + bcol] = acc[i] + bv;
    }
  }
}

// -------- training-mode BatchNorm1d ------------------------------------------
__global__ void bn_reduce(const float* __restrict__ x, int rows, int C,
                          float* __restrict__ stats) {
  __shared__ float ssum[256], ssq[256];
  const int c = blockIdx.x;
  float s = 0.f, q = 0.f;
  for (int r = threadIdx.x; r < rows; r += blockDim.x) {
    const float v = x[(size_t)r * C + c];
    s += v; q += v * v;
  }
  ssum[threadIdx.x] = s; ssq[threadIdx.x] = q;
  __syncthreads();
  for (int off = 128; off > 0; off >>= 1) {
    if ((int)threadIdx.x < off) {
      ssum[threadIdx.x] += ssum[threadIdx.x + off];
      ssq[threadIdx.x]  += ssq[threadIdx.x + off];
    }
    __syncthreads();
  }
  if (threadIdx.x == 0) { stats[c] = ssum[0]; stats[C + c] = ssq[0]; }
}

__global__ void bn_apply(const float* __restrict__ x, float* __restrict__ y,
                         int rows, int C, const float* __restrict__ stats,
                         const float* __restrict__ g, const float* __restrict__ b,
                         int relu) {
  const int total = rows * C;
  for (int i = blockIdx.x * blockDim.x + threadIdx.x; i < total;
       i += gridDim.x * blockDim.x) {
    const int c = i % C;
    const float mu  = stats[c] / rows;
    const float var = stats[C + c] / rows - mu * mu;
    float v = (x[i] - mu) * rsqrtf(var + 1e-5f) * g[c] + b[c];
    if (relu) v = fmaxf(v, 0.f);
    y[i] = v;
  }
}

// -------- global_add_pool ----------------------------------------------------
__global__ void pool_add(const float* __restrict__ h, const int* __restrict__ batch,
                         float* __restrict__ gp) {
  const int total = N_NODES * 72;
  for (int i = blockIdx.x * blockDim.x + threadIdx.x; i < total;
       i += gridDim.x * blockDim.x) {
    const int n = i / 72, c = i % 72;
    atomicAdd(&gp[(size_t)batch[n] * 72 + c], h[i]);
  }
}

// -------- final assembly: relu(c1/c2)+UNITSCALE, tanh(kij) -------------------
__global__ void assemble(const float* __restrict__ m1o, const float* __restrict__ m2o,
                         const float* __restrict__ m3o, float* __restrict__ out) {
  const int idx = blockIdx.x * blockDim.x + threadIdx.x;
  if (idx >= N_GRAPHS * 20) return;
  const int g = idx / 20, rem = idx % 20, row = rem / 10, col = rem % 10;
  const float US[7] = {1.f, 1.f, 100.f, 0.01f, 1000.f, 0.f, 1e-5f};
  float v;
  if (col < 7) {
    const float c = (row == 0) ? m2o[g * 7 + col] : m3o[g * 7 + col];
    v = fmaxf(c, 0.f) + US[col];
  } else {
    v = tanhf(m1o[g * 3 + (col - 7)]);
  }
  out[idx] = v;
}

// ============================================================================
static inline int cdiv(int a, int b) { return (a + b - 1) / b; }

extern "C" void kernel_launch(void* const* d_in, const int* in_sizes, int n_in,
                              void* d_out, int out_size, void* d_ws, size_t ws_size,
                              hipStream_t stream) {
  (void)in_sizes; (void)n_in; (void)out_size; (void)ws_size;

  // ---- inputs (setup_inputs order) ----
  const float* x   = (const float*)d_in[0];
  const float* ea  = (const float*)d_in[1];
  const int*   ei  = (const int*)d_in[2];
  const int*   bat = (const int*)d_in[3];
  const int* src = ei;                 // edge_index[0]
  const int* dst = ei + N_EDGES;       // edge_index[1]

  const float* c0_ew = (const float*)d_in[4];   const float* c0_eb = (const float*)d_in[5];
  const float* c0_prw = (const float*)d_in[6];  const float* c0_prb = (const float*)d_in[7];
  const float* c0_pow = (const float*)d_in[8];  const float* c0_pob = (const float*)d_in[9];
  const float* c0_lw = (const float*)d_in[10];  const float* c0_lb = (const float*)d_in[11];
  const float* c0_g  = (const float*)d_in[12];  const float* c0_be = (const float*)d_in[13];
  const float* cs_ew = (const float*)d_in[14];  const float* cs_eb = (const float*)d_in[15];
  const float* cs_prw = (const float*)d_in[16]; const float* cs_prb = (const float*)d_in[17];
  const float* cs_pow = (const float*)d_in[18]; const float* cs_pob = (const float*)d_in[19];
  const float* cs_lw = (const float*)d_in[20];  const float* cs_lb = (const float*)d_in[21];
  const float* cs_g  = (const float*)d_in[22];  const float* cs_be = (const float*)d_in[23];
  const float* hw[3][10];
  for (int m = 0; m < 3; ++m)
    for (int j = 0; j < 10; ++j) hw[m][j] = (const float*)d_in[24 + m * 10 + j];

  // AVG_DEG_LOG
  const double hist[7] = {67167, 3157428, 5106064, 885236, 453935, 0, 11152};
  double sl = 0.0, tt = 0.0;
  for (int i = 0; i < 7; ++i) { sl += log((double)i + 1.0) * hist[i]; tt += hist[i]; }
  const float avg_log = (float)(sl / tt);

  // ---- workspace carve-up (floats) ----
  float* W = (float*)d_ws;
  size_t off = 0;
  auto alloc = [&](size_t n) { float* p = W + off; off += n; return p; };
  float* Hbuf = alloc((size_t)N_EDGES * 216);   // gathered edge features
  float* msg  = alloc((size_t)N_EDGES * 288);   // per-tower messages
  float* s1   = alloc((size_t)N_NODES * 288);
  float* s2   = alloc((size_t)N_NODES * 288);
  float* mnb  = alloc((size_t)N_NODES * 288);
  float* mxb  = alloc((size_t)N_NODES * 288);
  float* deg  = alloc((size_t)N_NODES);
  float* hA   = alloc((size_t)N_NODES * 72);    // persistent node features
  float* hB   = alloc((size_t)N_NODES * 72);    // post-GEMM output
  float* hC   = alloc((size_t)N_NODES * 72);    // linear output (pre-BN)
  float* Wpk  = alloc((size_t)216 * 288);       // repacked pre-MLP weight
  float* stats= alloc(2 * 72);
  float* gp   = alloc((size_t)N_GRAPHS * 72);
  float* t50  = alloc((size_t)N_GRAPHS * 50);
  float* t25  = alloc((size_t)N_GRAPHS * 25);
  float* m1o  = alloc((size_t)N_GRAPHS * 3);
  float* m2o  = alloc((size_t)N_GRAPHS * 7);
  float* m3o  = alloc((size_t)N_GRAPHS * 7);

  // ---- one PNA conv layer + BN + relu ----
  auto run_layer = [&](const float* h_in, int F,
                       const float* ew, const float* eb, const float* prw,
                       const float* prb, const float* powW, const float* pob,
                       const float* lw, const float* lb,
                       const float* gamma, const float* beta, float* h_out) {
    const int K3 = 3 * F, TF = TOWERS * F, F13 = 13 * F;
    gather_edges<<<4096, 256, 0, stream>>>(h_in, F, ea, src, dst, ew, eb, Hbuf);
    pack_prw<<<256, 256, 0, stream>>>(prw, Wpk, F);
    dim3 g1(cdiv(N_EDGES, 16), cdiv(TF, 16));
    gemm_f32_wmma<<<g1, 32, 0, stream>>>(Hbuf, K3, Wpk, TF, prb, msg, TF,
                                         N_EDGES, TF, K3, 0);
    fill_agg<<<4096, 256, 0, stream>>>(s1, s2, mnb, mxb, deg, N_NODES * TF);
    deg_kernel<<<cdiv(N_EDGES, 256), 256, 0, stream>>>(dst, deg);
    scatter_agg<<<4096, 256, 0, stream>>>(msg, dst, s1, s2, mnb, mxb, TF);
    dim3 g2(cdiv(N_NODES, 16), TOWERS);
    pna_post<<<g2, 64, (size_t)16 * F13 * sizeof(float), stream>>>(
        h_in, F, s1, s2, mnb, mxb, deg, powW, pob, hB, avg_log);
    dim3 g3(cdiv(N_NODES, 16), cdiv(72, 16));
    gemm_f32_wmma<<<g3, 32, 0, stream>>>(hB, 72, lw, 72, lb, hC, 72,
                                         N_NODES, 72, 72, 0);
    bn_reduce<<<72, 256, 0, stream>>>(hC, N_NODES, 72, stats);
    bn_apply<<<4096, 256, 0, stream>>>(hC, h_out, N_NODES, 72, stats, gamma, beta, 1);
  };

  // layer 0 (F_in = 9) reads x directly
  run_layer(x, 9, c0_ew, c0_eb, c0_prw, c0_prb, c0_pow, c0_pob,
            c0_lw, c0_lb, c0_g, c0_be, hA);
  // layers 1..3 (F_in = 72), weights sliced from stacked arrays
  for (int i = 0; i < 3; ++i) {
    run_layer(hA, 72,
              cs_ew  + (size_t)i * 3 * 72,      cs_eb  + (size_t)i * 72,
              cs_prw + (size_t)i * TOWERS * 216 * 72,
              cs_prb + (size_t)i * TOWERS * 72,
              cs_pow + (size_t)i * TOWERS * 936 * 18,
              cs_pob + (size_t)i * TOWERS * 18,
              cs_lw  + (size_t)i * 72 * 72,     cs_lb  + (size_t)i * 72,
              cs_g   + (size_t)i * 72,          cs_be  + (size_t)i * 72, hA);
  }

  // global_add_pool
  fill_zero<<<128, 256, 0, stream>>>(gp, N_GRAPHS * 72);
  pool_add<<<4096, 256, 0, stream>>>(hA, bat, gp);

  // MLP heads: Linear->BN->relu, Linear->BN->relu, Linear
  auto run_head = [&](const float* const* w, int outd, float* outbuf) {
    dim3 ga(cdiv(N_GRAPHS, 16), cdiv(50, 16));
    gemm_f32_wmma<<<ga, 32, 0, stream>>>(gp, 72, w[0], 50, w[1], t50, 50,
                                         N_GRAPHS, 50, 72, 0);
    bn_reduce<<<50, 256, 0, stream>>>(t50, N_GRAPHS, 50, stats);
    bn_apply<<<128, 256, 0, stream>>>(t50, t50, N_GRAPHS, 50, stats, w[2], w[3], 1);
    dim3 gb(cdiv(N_GRAPHS, 16), cdiv(25, 16));
    gemm_f32_wmma<<<gb, 32, 0, stream>>>(t50, 50, w[4], 25, w[5], t25, 25,
                                         N_GRAPHS, 25, 50, 0);
    bn_reduce<<<25, 256, 0, stream>>>(t25, N_GRAPHS, 25, stats);
    bn_apply<<<64, 256, 0, stream>>>(t25, t25, N_GRAPHS, 25, stats, w[6], w[7], 1);
    dim3 gc(cdiv(N_GRAPHS, 16), cdiv(outd, 16));
    gemm_f32_wmma<<<gc, 32, 0, stream>>>(t25, 25, w[8], outd, w[9], outbuf, outd,
                                         N_GRAPHS, outd, 25, 0);
  };
  run_head(hw[0], 3, m1o);   // kij (tanh applied in assemble)
  run_head(hw[1], 7, m2o);   // c1
  run_head(hw[2], 7, m3o);   // c2

  assemble<<<cdiv(N_GRAPHS * 20, 256), 256, 0, stream>>>(m1o, m2o, m3o, (float*)d_out);
}